// IterativeRefiner_5935644803528
// MI455X (gfx1250) — compile-verified
//
#include <hip/hip_runtime.h>

// ---------------------------------------------------------------------------
// IterativeRefiner for MI455X (gfx1250): fp32 WMMA (16x16x4) GEMMs with
// async-to-LDS double-buffered pipelines. B=4, N=256, D_IN=128, D=256, T=3.
// ---------------------------------------------------------------------------

typedef __attribute__((ext_vector_type(2))) float v2f;
typedef __attribute__((ext_vector_type(8))) float v8f;
typedef __attribute__((ext_vector_type(4))) int   v4i;

#define BN_ 256     // graph nodes
#define NB_ 4       // batch
#define TT_ 3

// ---- CDNA5 async global->LDS copy (guarded; falls back to sync copy) ------
#if defined(__AMDGCN__) && __has_builtin(__builtin_amdgcn_global_load_async_to_lds_b128)
#define ASYNC_LDS 1
typedef __attribute__((address_space(1))) v4i gv4i_t;   // global int4
typedef __attribute__((address_space(3))) v4i sv4i_t;   // LDS int4
#endif

__device__ __forceinline__ void ld16(void* lds, const void* g) {
#ifdef ASYNC_LDS
    __builtin_amdgcn_global_load_async_to_lds_b128(
        (gv4i_t*)(g), (sv4i_t*)(lds), /*offset=*/0, /*cpol=*/0);
#else
    *(float4*)lds = *(const float4*)g;
#endif
}

__device__ __forceinline__ void async_wait0() {
#ifdef ASYNC_LDS
#if __has_builtin(__builtin_amdgcn_s_wait_asynccnt)
    __builtin_amdgcn_s_wait_asynccnt(0);
#else
    asm volatile("s_wait_asynccnt 0x0" ::: "memory");
#endif
#endif
}

__device__ __forceinline__ v8f wmma4(v2f a, v2f b, v8f c) {
    // D = A(16x4 f32) * B(4x16 f32) + C(16x16 f32)
    return __builtin_amdgcn_wmma_f32_16x16x4_f32(
        /*neg_a=*/false, a, /*neg_b=*/false, b,
        /*c_mod=*/(short)0, c, /*reuse_a=*/false, /*reuse_b=*/false);
}

// ---------------------------------------------------------------------------
// Generic WMMA GEMM: C[M,N] = A[M,K] @ B[K,N] (+bias +rowterm +residual, relu)
// M mult of 64, N mult of 64 (here N==256), K mult of 32.
// Block: 256 threads (8 waves), tile 64x64; wave owns a 16x32 strip.
// Double-buffered LDS (BK=32) fed by async global->LDS b128 copies.
// strideAb/strideBb: per-256-row batch offsets (0 = shared operand).
// rowterm: [4][256] indexed (row/256, col). outT: outT[b][col][row%256].
// ---------------------------------------------------------------------------
#define BK_ 32
#define ASTR_ 36   // padded A row stride (keeps 16B chunks, 8B frag align)

__global__ __launch_bounds__(256) void gemm_wmma_f32(
    const float* __restrict__ A, int lda, int strideAb,
    const float* __restrict__ B, int ldb, int strideBb,
    float* __restrict__ C, int ldc,
    const float* __restrict__ bias,
    const float* __restrict__ rowterm,
    const float* __restrict__ residual,
    float* __restrict__ outT,
    int relu, int K)
{
    __shared__ float As[2][64 * ASTR_];
    __shared__ float Bs[2][32 * 64];

    const int tid  = threadIdx.x;
    const int lane = tid & 31;
    const int wave = tid >> 5;
    const int lm   = lane & 15;     // N-col / M-row within 16
    const int kh   = lane >> 4;     // half selector (K-pairs / M+8)
    const int wm   = wave & 3;      // wave M-tile (4 x 16 rows)
    const int wn   = wave >> 2;     // wave N-strip (2 x 32 cols)

    const int m0 = blockIdx.y * 64;
    const int n0 = blockIdx.x * 64;
    const int batch = m0 >> 8;

    // cooperative copy indices (each thread moves 2x16B of A, 2x16B of B)
    const int arow  = tid >> 2;           // 0..63
    const int acol0 = (tid & 3) * 4;      // 0,4,8,12 (+16)
    const int brow0 = tid >> 4;           // 0..15 (+16)
    const int bcol  = (tid & 15) * 4;     // 0..60

    const int grow = m0 + arow;
    long aRowBase;
    if (strideAb) aRowBase = (long)batch * strideAb + (long)(grow & 255) * lda;
    else          aRowBase = (long)grow * lda;
    const long bBatchBase = strideBb ? (long)batch * strideBb : 0l;

    auto issue = [&](int kk, int bb) {
#pragma unroll
        for (int i = 0; i < 2; ++i) {
            const int ac = acol0 + i * 16;
            ld16(&As[bb][arow * ASTR_ + ac], A + aRowBase + kk + ac);
            const int br = brow0 + i * 16;
            ld16(&Bs[bb][br * 64 + bcol],
                 B + bBatchBase + (long)(kk + br) * ldb + n0 + bcol);
        }
    };

    v8f acc0 = {}; v8f acc1 = {};

    issue(0, 0);
    async_wait0();
    __syncthreads();

    int buf = 0;
    for (int kk = 0; kk < K; kk += BK_) {
        const bool more = (kk + BK_) < K;
        if (more) issue(kk + BK_, buf ^ 1);     // prefetch next tile (async)

        const int aBase = (wm * 16 + lm) * ASTR_;
        const int cCol  = wn * 32 + lm;
#pragma unroll
        for (int ks = 0; ks < 8; ++ks) {
            const int k0 = ks * 4 + kh * 2;     // lane's K-pair start
            v2f a = *(const v2f*)&As[buf][aBase + k0];
            v2f b0, b1;
            b0.x = Bs[buf][(k0 + 0) * 64 + cCol];
            b0.y = Bs[buf][(k0 + 1) * 64 + cCol];
            b1.x = Bs[buf][(k0 + 0) * 64 + cCol + 16];
            b1.y = Bs[buf][(k0 + 1) * 64 + cCol + 16];
            acc0 = wmma4(a, b0, acc0);
            acc1 = wmma4(a, b1, acc1);
        }
        if (more) async_wait0();
        __syncthreads();
        buf ^= 1;
    }

    // epilogue
    union { v8f v; float f[8]; } u0, u1;
    u0.v = acc0; u1.v = acc1;
#pragma unroll
    for (int j = 0; j < 2; ++j) {
        const float* uf = j ? u1.f : u0.f;
        const int col = n0 + wn * 32 + j * 16 + lm;
#pragma unroll
        for (int r = 0; r < 8; ++r) {
            const int row = m0 + wm * 16 + kh * 8 + r;
            float v = uf[r];
            if (bias)     v += bias[col];
            if (rowterm)  v += rowterm[(row >> 8) * 256 + col];
            if (residual) v += residual[(long)row * ldc + col];
            if (relu)     v = fmaxf(v, 0.f);
            C[(long)row * ldc + col] = v;
            if (outT) outT[(long)(row >> 8) * 65536 + (long)col * 256 + (row & 255)] = v;
        }
    }
}

// ---------------------------------------------------------------------------
// Fused MLPAdjacency: block (e,b), thread n:
//   inc[b,e,n] = sigmoid( sum_d relu(xp[b,n,d]+xp[b,e,d]+inc0[b,e,n]*wi[d]+bi[d]) * ws[d] + bs )
// xpT[b][d][n] streamed through a 2x16KB LDS double buffer via async copies.
// Also emits incT[b][n][e] for the following upd GEMM.
// ---------------------------------------------------------------------------
__global__ __launch_bounds__(256) void adjacency_fused(
    const float* __restrict__ xp, const float* __restrict__ xpT,
    const float* __restrict__ incprev,
    const float* __restrict__ wi, const float* __restrict__ bi,
    const float* __restrict__ wsv, const float* __restrict__ bsv,
    float* __restrict__ incout, float* __restrict__ incTout)
{
    __shared__ float xe[256], wiL[256], biL[256], wsL[256];
    __shared__ float xt[2][16 * 256];     // 16 d-rows per buffer

    const int e = blockIdx.x, b = blockIdx.y, n = threadIdx.x;
    const int tid = threadIdx.x;
    const long beBase = (long)b * 65536 + (long)e * 256;
    const float* xtb = xpT + (long)b * 65536;

    xe[n]  = xp[beBase + n];
    wiL[n] = wi[n]; biL[n] = bi[n]; wsL[n] = wsv[n];

    auto issue = [&](int c, int bb) {   // copy d-rows [c*16, c*16+16) of xpT[b]
#pragma unroll
        for (int i = 0; i < 4; ++i) {
            const int dl  = (tid >> 6) + i * 4;   // 0..15
            const int col = (tid & 63) * 4;
            ld16(&xt[bb][dl * 256 + col], xtb + (long)(c * 16 + dl) * 256 + col);
        }
    };

    issue(0, 0);
    async_wait0();
    __syncthreads();

    const float incv = incprev[beBase + n];
    float acc = 0.f;
    int buf = 0;
    for (int c = 0; c < 16; ++c) {
        const bool more = (c + 1) < 16;
        if (more) issue(c + 1, buf ^ 1);
#pragma unroll
        for (int dd = 0; dd < 16; ++dd) {
            const int d = c * 16 + dd;
            float t = xt[buf][dd * 256 + n] + xe[d] + incv * wiL[d] + biL[d];
            acc = fmaf(fmaxf(t, 0.f), wsL[d], acc);
        }
        if (more) { async_wait0(); }
        __syncthreads();
        buf ^= 1;
    }

    const float z  = acc + bsv[0];
    const float sg = 1.f / (1.f + __expf(-z));
    incout[beBase + n] = sg;
    incTout[(long)b * 65536 + (long)n * 256 + e] = sg;
}

// ---------------------------------------------------------------------------
// LayerNorm over the 768-wide concat [x_in | n_t | upd], one block per (b,n).
// ---------------------------------------------------------------------------
__global__ __launch_bounds__(256) void ln_concat768(
    const float* __restrict__ xin, const float* __restrict__ nt,
    const float* __restrict__ upd,
    const float* __restrict__ g, const float* __restrict__ be,
    float* __restrict__ h)
{
    __shared__ float sS[256], sQ[256];
    const int row = blockIdx.x, tid = threadIdx.x;
    const long base = (long)row * 256;
    const float a = xin[base + tid], b = nt[base + tid], c = upd[base + tid];
    sS[tid] = a + b + c;
    sQ[tid] = a * a + b * b + c * c;
    __syncthreads();
    for (int off = 128; off > 0; off >>= 1) {
        if (tid < off) { sS[tid] += sS[tid + off]; sQ[tid] += sQ[tid + off]; }
        __syncthreads();
    }
    const float mu  = sS[0] * (1.f / 768.f);
    const float var = sQ[0] * (1.f / 768.f) - mu * mu;
    const float rs  = rsqrtf(var + 1e-5f);
    const long hb = (long)row * 768;
    h[hb + tid]       = (a - mu) * rs * g[tid]       + be[tid];
    h[hb + 256 + tid] = (b - mu) * rs * g[256 + tid] + be[256 + tid];
    h[hb + 512 + tid] = (c - mu) * rs * g[512 + tid] + be[512 + tid];
}

// LayerNorm over 256, one block per (b,n):  out = LN(z) * g + be
__global__ __launch_bounds__(256) void ln_256(
    const float* __restrict__ z,
    const float* __restrict__ g, const float* __restrict__ be,
    float* __restrict__ out)
{
    __shared__ float sS[256], sQ[256];
    const int row = blockIdx.x, tid = threadIdx.x;
    const float a = z[(long)row * 256 + tid];
    sS[tid] = a; sQ[tid] = a * a;
    __syncthreads();
    for (int off = 128; off > 0; off >>= 1) {
        if (tid < off) { sS[tid] += sS[tid + off]; sQ[tid] += sQ[tid + off]; }
        __syncthreads();
    }
    const float mu  = sS[0] * (1.f / 256.f);
    const float var = sQ[0] * (1.f / 256.f) - mu * mu;
    const float rs  = rsqrtf(var + 1e-5f);
    out[(long)row * 256 + tid] = (a - mu) * rs * g[tid] + be[tid];
}

// m[b][j] = mean_n X[b][n][j];  grid (B, F/256)
__global__ __launch_bounds__(256) void mean_over_n(
    const float* __restrict__ X, int F, float* __restrict__ m)
{
    const int b = blockIdx.x;
    const int j = blockIdx.y * 256 + threadIdx.x;
    const long base = (long)b * 256 * F + j;
    float acc = 0.f;
    for (int n = 0; n < 256; ++n) acc += X[base + (long)n * F];
    m[b * F + j] = acc * (1.f / 256.f);
}

// outv[b][o] = sum_j mvec[b][j] * W[j][o] + bias[o];  grid (B), 256 threads
__global__ __launch_bounds__(256) void gemv_rowterm(
    const float* __restrict__ mvec, int K,
    const float* __restrict__ W, const float* __restrict__ bias,
    float* __restrict__ outv)
{
    const int r = blockIdx.x, o = threadIdx.x;
    float acc = bias[o];
    for (int j = 0; j < K; ++j) acc = fmaf(mvec[r * K + j], W[j * 256 + o], acc);
    outv[r * 256 + o] = acc;
}

// ---------------------------------------------------------------------------
extern "C" void kernel_launch(void* const* d_in, const int* in_sizes, int n_in,
                              void* d_out, int out_size, void* d_ws, size_t ws_size,
                              hipStream_t stream) {
    (void)in_sizes; (void)n_in; (void)out_size; (void)ws_size;
    const float* inputs = (const float*)d_in[0];   // [4,256,128]
    const float* v_t    = (const float*)d_in[1];   // [4,256,256]
    const float* i_t    = (const float*)d_in[2];   // [4,256,256]
    const float* Wp     = (const float*)d_in[3];
    const float* bp     = (const float*)d_in[4];
    const float* Wx     = (const float*)d_in[5];
    const float* bx     = (const float*)d_in[6];
    const float* wi     = (const float*)d_in[7];
    const float* bi     = (const float*)d_in[8];
    const float* wsv    = (const float*)d_in[9];
    const float* bsv    = (const float*)d_in[10];
    const float* g_pre  = (const float*)d_in[11];
    const float* be_pre = (const float*)d_in[12];
    const float* g_n    = (const float*)d_in[13];
    const float* be_n   = (const float*)d_in[14];
    const float* W1g    = (const float*)d_in[15];
    const float* W1l    = (const float*)d_in[16];
    const float* b1     = (const float*)d_in[17];
    const float* W2g    = (const float*)d_in[18];
    const float* W2l    = (const float*)d_in[19];
    const float* b2     = (const float*)d_in[20];

    float* out = (float*)d_out;                    // [3*BNN | BNN | BNN]
    const long BNN = (long)NB_ * BN_ * BN_;        // 262144

    float* w    = (float*)d_ws;
    float* x_in = w;                 // 262144
    float* xp   = w + 262144;
    float* xpT  = w + 524288;
    float* incT = w + 786432;
    float* upd  = w + 1048576;
    float* h    = w + 1310720;       // 786432
    float* h1   = w + 2097152;
    float* ztmp = w + 2359296;
    float* mbuf = w + 2621440;       // 4*768
    float* m1   = w + 2624512;       // 4*256
    float* mt1  = w + 2625536;       // 4*256
    float* mt2  = w + 2626560;       // 4*256
    float* nt0  = w + 2627584;
    float* nt1  = w + 2889728;

    const dim3 gGemm(4, 16);   // N/64, M/64
    const dim3 blk(256);

    // x_in = inputs @ Wp + bp
    gemm_wmma_f32<<<gGemm, blk, 0, stream>>>(inputs, 128, 0, Wp, 256, 0,
        x_in, 256, bp, nullptr, nullptr, nullptr, 0, 128);

    for (int t = 0; t < TT_; ++t) {
        const float* nt  = (t == 0) ? v_t : ((t == 1) ? nt0 : nt1);
        float*       ntw = (t & 1) ? nt1 : nt0;     // t0->nt0, t1->nt1, t2->nt0
        const float* incprev = (t == 0) ? i_t : (out + (long)(t - 1) * BNN);
        float*       inccur  = out + (long)t * BNN;

        // xp = nt @ Wx + bx   (also emit xpT[b][d][n])
        gemm_wmma_f32<<<gGemm, blk, 0, stream>>>(nt, 256, 0, Wx, 256, 0,
            xp, 256, bx, nullptr, nullptr, xpT, 0, 256);

        // inc = sigmoid(relu(...) . ws + bs)  (+ incT)
        adjacency_fused<<<dim3(256, 4), blk, 0, stream>>>(
            xp, xpT, incprev, wi, bi, wsv, bsv, inccur, incT);

        // upd[b] = incT[b] @ nt[b]   (batched)
        gemm_wmma_f32<<<gGemm, blk, 0, stream>>>(incT, 256, 65536, nt, 256, 65536,
            upd, 256, nullptr, nullptr, nullptr, nullptr, 0, 256);

        // h = LN768([x_in | nt | upd])
        ln_concat768<<<1024, blk, 0, stream>>>(x_in, nt, upd, g_pre, be_pre, h);

        // m = mean_n(h);  mt1 = m @ W1l + b1
        mean_over_n<<<dim3(4, 3), blk, 0, stream>>>(h, 768, mbuf);
        gemv_rowterm<<<4, blk, 0, stream>>>(mbuf, 768, W1l, b1, mt1);

        // h1 = relu(h @ W1g + mt1[b])
        gemm_wmma_f32<<<gGemm, blk, 0, stream>>>(h, 768, 0, W1g, 256, 0,
            h1, 256, nullptr, mt1, nullptr, nullptr, 1, 768);

        // m1 = mean_n(h1);  mt2 = m1 @ W2l + b2
        mean_over_n<<<dim3(4, 1), blk, 0, stream>>>(h1, 256, m1);
        gemv_rowterm<<<4, blk, 0, stream>>>(m1, 256, W2l, b2, mt2);

        // z = h1 @ W2g + mt2[b] + nt ;  nt_next = LN256(z)
        gemm_wmma_f32<<<gGemm, blk, 0, stream>>>(h1, 256, 0, W2g, 256, 0,
            ztmp, 256, nullptr, mt2, nt, nullptr, 0, 256);
        ln_256<<<1024, blk, 0, stream>>>(ztmp, g_n, be_n, ntw);
    }

    // outputs: preds at out[0..3*BNN); n_t = nt0 (written by t=2); inc = preds[2]
    (void)hipMemcpyAsync(out + TT_ * BNN, nt0, BNN * sizeof(float),
                         hipMemcpyDeviceToDevice, stream);
    (void)hipMemcpyAsync(out + (TT_ + 1) * BNN, out + (TT_ - 1) * BNN, BNN * sizeof(float),
                         hipMemcpyDeviceToDevice, stream);
}